// andps_images_34110630264834
// MI455X (gfx1250) — compile-verified
//
#include <hip/hip_runtime.h>
#include <hip/hip_bf16.h>
#include <math.h>

typedef float v2f __attribute__((ext_vector_type(2)));
typedef float v8f __attribute__((ext_vector_type(8)));

#define N_CLASSES 26
#define IMG 68
#define BATCH 8192
#define FLAT 1960
#define HID 500
#define NEXP 64
#define FC1_IN 1962   // FLAT + 2
#define HSTR 516      // padded LDS row stride (516 % 64 banks = 4 -> conflict-free b64 loads)

// ---------------------------------------------------------------------------
// Kernel 1: per-class gating CNN. One block per class.
// conv1(1->5,5x5,VALID) + relu + maxpool2 -> [5,32,32]
// conv2(5->10,5x5,VALID) + relu + maxpool2 -> [10,14,14] -> hcnn[c][1960]
// ---------------------------------------------------------------------------
__global__ __launch_bounds__(256) void cnn_kernel(
    const float* __restrict__ images,
    const float* __restrict__ c1w, const float* __restrict__ c1b,
    const float* __restrict__ c2w, const float* __restrict__ c2b,
    float* __restrict__ hcnn)
{
    __shared__ float s_img[IMG * IMG];     // 18496 B
    __shared__ float s_h1[5 * 32 * 32];    // 20480 B
    __shared__ float s_w1[125], s_b1[5], s_w2[1250], s_b2[10];

    const int c = blockIdx.x, t = threadIdx.x;
    for (int i = t; i < IMG * IMG; i += 256) s_img[i] = images[c * IMG * IMG + i];
    for (int i = t; i < 125;  i += 256) s_w1[i] = c1w[i];
    for (int i = t; i < 5;    i += 256) s_b1[i] = c1b[i];
    for (int i = t; i < 1250; i += 256) s_w2[i] = c2w[i];
    for (int i = t; i < 10;   i += 256) s_b2[i] = c2b[i];
    __syncthreads();

    // conv1 + relu + pool
    for (int o = t; o < 5 * 32 * 32; o += 256) {
        int ch = o >> 10, py = (o >> 5) & 31, px = o & 31;
        float best = 0.f;   // relu outputs are >= 0
        for (int dy = 0; dy < 2; ++dy)
        for (int dx = 0; dx < 2; ++dx) {
            int y = 2 * py + dy, x = 2 * px + dx;
            float acc = s_b1[ch];
            for (int ky = 0; ky < 5; ++ky) {
                const float* ir = &s_img[(y + ky) * IMG + x];
                const float* wr = &s_w1[ch * 25 + ky * 5];
                #pragma unroll
                for (int kx = 0; kx < 5; ++kx) acc += ir[kx] * wr[kx];
            }
            best = fmaxf(best, fmaxf(acc, 0.f));
        }
        s_h1[o] = best;
    }
    __syncthreads();

    // conv2 + relu + pool
    for (int o = t; o < 10 * 14 * 14; o += 256) {
        int ch = o / 196, r = o % 196, py = r / 14, px = r % 14;
        float best = 0.f;
        for (int dy = 0; dy < 2; ++dy)
        for (int dx = 0; dx < 2; ++dx) {
            int y = 2 * py + dy, x = 2 * px + dx;
            float acc = s_b2[ch];
            for (int ic = 0; ic < 5; ++ic) {
                const float* hp = &s_h1[ic * 1024 + y * 32 + x];
                const float* wp = &s_w2[(ch * 5 + ic) * 25];
                for (int ky = 0; ky < 5; ++ky) {
                    #pragma unroll
                    for (int kx = 0; kx < 5; ++kx) acc += hp[ky * 32 + kx] * wp[ky * 5 + kx];
                }
            }
            best = fmaxf(best, fmaxf(acc, 0.f));
        }
        hcnn[c * FLAT + o] = best;
    }
}

// ---------------------------------------------------------------------------
// Kernel 2: per-class fc1 partials  P[c][j] = fc1_w[j,0:1960].hcnn[c] + fc1_b[j]
// One wave32 per output, shuffle reduction.  grid = (63, 26), block = 256.
// ---------------------------------------------------------------------------
__global__ __launch_bounds__(256) void pcls_kernel(
    const float* __restrict__ hcnn, const float* __restrict__ fc1w,
    const float* __restrict__ fc1b, float* __restrict__ P)
{
    const int c = blockIdx.y;
    const int wave = threadIdx.x >> 5, lane = threadIdx.x & 31;
    const int j = blockIdx.x * 8 + wave;
    if (j >= HID) return;                      // wave-uniform branch
    const float* hr = &hcnn[c * FLAT];
    const float* wr = &fc1w[j * FC1_IN];
    float s = 0.f;
    for (int k = lane; k < FLAT; k += 32) s += hr[k] * wr[k];
    #pragma unroll
    for (int o = 16; o > 0; o >>= 1) s += __shfl_xor(s, o, 32);
    if (lane == 0) P[c * HID + j] = s + fc1b[j];
}

// ---------------------------------------------------------------------------
// Kernel 3: per-sample fc1 finish + fc2 via V_WMMA_F32_16X16X4_F32 + softmax
// + fused expert 2x2 weighted sum. 16 samples per block, 4 waves.
// ---------------------------------------------------------------------------
__global__ __launch_bounds__(128) void moe_kernel(
    const float* __restrict__ x, const int* __restrict__ cls_idx,
    const float* __restrict__ fc1w, const float* __restrict__ P,
    const float* __restrict__ fc2w, const float* __restrict__ fc2b,
    const float* __restrict__ A_B, const float* __restrict__ A_C,
    const float* __restrict__ x_tar, float* __restrict__ out)
{
    __shared__ float s_h[16 * HSTR];       // 33024 B
    __shared__ float s_logits[16 * NEXP];  //  4096 B
    __shared__ int   s_cls[16];
    __shared__ float s_x[16][2];

    const int t = threadIdx.x;
    const int s0 = blockIdx.x * 16;

    if (t < 16) {
        s_cls[t]  = cls_idx[s0 + t];
        s_x[t][0] = x[(s0 + t) * 2 + 0];
        s_x[t][1] = x[(s0 + t) * 2 + 1];
    }
    __syncthreads();

    // Phase B: h[s][j] = relu(P[cls[s]][j] + w_x0[j]*x0 + w_x1[j]*x1)
    for (int idx = t; idx < 16 * HID; idx += 128) {
        int s = idx / HID, j = idx % HID;
        float v = P[s_cls[s] * HID + j]
                + fc1w[j * FC1_IN + FLAT]     * s_x[s][0]
                + fc1w[j * FC1_IN + FLAT + 1] * s_x[s][1];
        s_h[s * HSTR + j] = fmaxf(v, 0.f);
    }
    __syncthreads();

    // Phase C: logits[16x64] = h[16x500] @ fc2_w^T[500x64], one 16x16 tile/wave.
    // A 16x4 f32 layout: lanes 0-15 -> K=0,1 ; lanes 16-31 -> K=2,3 (B mirrored).
    const int wave = t >> 5, lane = t & 31;
    const int half = lane >> 4;        // 0 | 1
    const int ml   = lane & 15;        // A row m / B col n
    const int kofs = half * 2;
    const int N    = wave * 16 + ml;

    const float* hrow = &s_h[ml * HSTR];
    const float* wcol = &fc2w[N * HID];    // B column N == fc2_w row N

    v8f acc = {};
    for (int kb = 0; kb < HID; kb += 4) {  // 125 WMMA steps, K = 500 exactly
        v2f a, b;
        a.x = hrow[kb + kofs];  a.y = hrow[kb + kofs + 1];
        b.x = wcol[kb + kofs];  b.y = wcol[kb + kofs + 1];
        acc = __builtin_amdgcn_wmma_f32_16x16x4_f32(
            /*neg_a=*/false, a, /*neg_b=*/false, b,
            /*c_mod=*/(short)0, acc, /*reuse_a=*/false, /*reuse_b=*/false);
    }

    const float bias = fc2b[N];
    #pragma unroll
    for (int r = 0; r < 8; ++r) {
        int M = r + 8 * half;              // C/D layout: VGPR r -> M=r (lanes 0-15), M=r+8 (16-31)
        s_logits[M * NEXP + N] = acc[r] + bias;
    }
    __syncthreads();

    // Phase D: softmax over 64 experts + fused Sum_n w_n * A_n @ (x_tar - x)
    if (t < 16) {
        const int s = t;
        float mx = -1e30f;
        for (int n = 0; n < NEXP; ++n) mx = fmaxf(mx, s_logits[s * NEXP + n]);
        float se = 0.f, m00 = 0.f, m01 = 0.f, m10 = 0.f, m11 = 0.f;
        for (int n = 0; n < NEXP; ++n) {
            float e = expf(s_logits[s * NEXP + n] - mx);
            se  += e;
            m00 += e * (A_B[n * 4 + 0] + A_C[n * 4 + 0]);
            m01 += e * (A_B[n * 4 + 1] + A_C[n * 4 + 1]);
            m10 += e * (A_B[n * 4 + 2] + A_C[n * 4 + 2]);
            m11 += e * (A_B[n * 4 + 3] + A_C[n * 4 + 3]);
        }
        float d0 = x_tar[0] - s_x[s][0];
        float d1 = x_tar[1] - s_x[s][1];
        float inv = 1.f / se;
        out[(s0 + s) * 2 + 0] = (m00 * d0 + m01 * d1) * inv;
        out[(s0 + s) * 2 + 1] = (m10 * d0 + m11 * d1) * inv;
    }
}

// ---------------------------------------------------------------------------
extern "C" void kernel_launch(void* const* d_in, const int* in_sizes, int n_in,
                              void* d_out, int out_size, void* d_ws, size_t ws_size,
                              hipStream_t stream)
{
    const float* x      = (const float*)d_in[0];
    const int*   cls    = (const int*)  d_in[1];
    const float* images = (const float*)d_in[2];
    const float* c1w    = (const float*)d_in[3];
    const float* c1b    = (const float*)d_in[4];
    const float* c2w    = (const float*)d_in[5];
    const float* c2b    = (const float*)d_in[6];
    const float* f1w    = (const float*)d_in[7];
    const float* f1b    = (const float*)d_in[8];
    const float* f2w    = (const float*)d_in[9];
    const float* f2b    = (const float*)d_in[10];
    const float* AB     = (const float*)d_in[11];
    const float* AC     = (const float*)d_in[12];
    const float* xt     = (const float*)d_in[13];
    float* out = (float*)d_out;

    float* hcnn = (float*)d_ws;                    // [26 * 1960]
    float* P    = hcnn + N_CLASSES * FLAT;         // [26 * 500]

    cnn_kernel<<<N_CLASSES, 256, 0, stream>>>(images, c1w, c1b, c2w, c2b, hcnn);
    pcls_kernel<<<dim3(63, N_CLASSES), 256, 0, stream>>>(hcnn, f1w, f1b, P);
    moe_kernel<<<BATCH / 16, 128, 0, stream>>>(x, cls, f1w, P, f2w, f2b, AB, AC, xt, out);
}